// model3_neural_11776800326273
// MI455X (gfx1250) — compile-verified
//
#include <hip/hip_runtime.h>
#include <math.h>

// Problem constants (match reference)
#define NT 4096      // tasks
#define NS 200       // support items
#define NQ 100       // query items
#define ND 32        // embedding dim
#define SPAD 208     // 13 tiles of 16
#define QPAD 112     // 7 tiles of 16
#define PITCH 33     // LDS row pitch (bank-conflict-free for stride-32 column reads)
#define LAM_U 0.01f
#define LOCAL_LR 0.01f

typedef float v2f __attribute__((ext_vector_type(2)));
typedef float v8f __attribute__((ext_vector_type(8)));

#define AS1 __attribute__((address_space(1)))
#define AS3 __attribute__((address_space(3)))

// ---- CDNA5 async global->LDS gather path (guarded; falls back to sync copy) ----
#if defined(__gfx1250__) && __has_builtin(__builtin_amdgcn_global_load_async_to_lds_b32)
#define HAVE_ASYNC_LDS 1
#else
#define HAVE_ASYNC_LDS 0
#endif

__device__ __forceinline__ void gather_dword(const float* __restrict__ gsrc,
                                             float* __restrict__ lds_dst) {
#if HAVE_ASYNC_LDS
  // global_load_async_to_lds_b32: per-lane 4B memory->LDS, tracked on ASYNCcnt,
  // no VGPR return -> fire-and-forget, one wait at the end of the gather.
  __builtin_amdgcn_global_load_async_to_lds_b32(
      (AS1 int*)(unsigned long long)gsrc,
      (AS3 int*)(unsigned)(unsigned long long)lds_dst,
      /*offset=*/0, /*cpol=*/0);
#else
  *lds_dst = *gsrc;
#endif
}

__device__ __forceinline__ void gather_wait() {
#if HAVE_ASYNC_LDS
#if __has_builtin(__builtin_amdgcn_s_wait_asynccnt)
  __builtin_amdgcn_s_wait_asynccnt(0);
#else
  asm volatile("s_wait_asynccnt 0" ::: "memory");
#endif
#endif
}

__device__ __forceinline__ float sigm(float x) { return 1.0f / (1.0f + expf(-x)); }

// Batched dot products z[s] = sum_d M[s][d]*u[d] for ntiles*16 rows, via
// V_WMMA_F32_16X16X4_F32. u is replicated across all 16 B-columns, so every
// column of D holds z; lanes with (lane&15)==0 extract rows per the C/D layout.
// A-fragment layout (16x4 f32): VGPR0: K=0 (lanes 0-15) / K=2 (lanes 16-31),
// VGPR1: K=1 / K=3.
__device__ __forceinline__ void wmma_z(const float* __restrict__ M, int ntiles,
                                       const float* __restrict__ u,
                                       float* __restrict__ z, int lane) {
  const int l15 = lane & 15;
  const int koffx = (lane >= 16) ? 2 : 0;
  const int koffy = (lane >= 16) ? 3 : 1;
  for (int tile = 0; tile < ntiles; ++tile) {
    const int s0 = tile * 16;
    v8f c = {0.f, 0.f, 0.f, 0.f, 0.f, 0.f, 0.f, 0.f};
#pragma unroll
    for (int k = 0; k < ND; k += 4) {
      v2f a, b;
      a.x = M[(s0 + l15) * PITCH + k + koffx];
      a.y = M[(s0 + l15) * PITCH + k + koffy];
      b.x = u[k + koffx];   // same value for every column N
      b.y = u[k + koffy];
      c = __builtin_amdgcn_wmma_f32_16x16x4_f32(false, a, false, b, (short)0, c,
                                                false, false);
    }
    if (l15 == 0) {
      const int base = s0 + ((lane >= 16) ? 8 : 0);
#pragma unroll
      for (int j = 0; j < 8; ++j) z[base + j] = c[j];
    }
  }
}

// Kernel 1: one wave (32 lanes) per task. Stages E_s/E_q into LDS once,
// runs MLP + 5 GD steps + query pass; emits per-task h, x, tmp, dpre1, loss.
__global__ void __launch_bounds__(32)
k1_per_task(const float* __restrict__ emb, const float* __restrict__ W1,
            const float* __restrict__ b1, const float* __restrict__ W2,
            const float* __restrict__ b2, const float* __restrict__ share,
            const float* __restrict__ supp_y, const float* __restrict__ query_y,
            const int* __restrict__ supp_itms, const int* __restrict__ query_itms,
            float* __restrict__ h_ws, float* __restrict__ x_ws,
            float* __restrict__ tmp_ws, float* __restrict__ dpre1_ws,
            float* __restrict__ loss_ws) {
  __shared__ float Es[SPAD * PITCH];
  __shared__ float Eq[QPAD * PITCH];
  __shared__ float zsh[SPAD];
  __shared__ float gsh[SPAD];
  __shared__ float ysup[NS];
  __shared__ float yqry[NQ];
  __shared__ float xsh[64];
  __shared__ float hsh[32];
  __shared__ float presh[32];
  __shared__ float ush[32];
  __shared__ float tmpsh[32];
  __shared__ float red[32];

  const int t = blockIdx.x;
  const int lane = threadIdx.x;

  // ---- gather embeddings into LDS (rows coalesced 128B, async fire-and-forget) ----
  const int* si = supp_itms + t * NS;
  for (int s = 0; s < NS; ++s) {
    gather_dword(&emb[(size_t)si[s] * ND + lane], &Es[s * PITCH + lane]);
  }
  const int* qi = query_itms + t * NQ;
  for (int q = 0; q < NQ; ++q) {
    gather_dword(&emb[(size_t)qi[q] * ND + lane], &Eq[q * PITCH + lane]);
  }
  for (int s = NS; s < SPAD; ++s) Es[s * PITCH + lane] = 0.f;
  for (int q = NQ; q < QPAD; ++q) Eq[q * PITCH + lane] = 0.f;
  for (int s = lane; s < NS; s += 32) ysup[s] = supp_y[t * NS + s];
  for (int q = lane; q < NQ; q += 32) yqry[q] = query_y[t * NQ + q];
  gather_wait();
  __syncthreads();

  // ---- pos_emb = mean of support slots 0,5,...,195 ; x = [pos_emb, share] ----
  float acc = 0.f;
#pragma unroll
  for (int i = 0; i < 40; ++i) acc += Es[(i * 5) * PITCH + lane];
  xsh[lane] = acc * (1.0f / 40.0f);
  xsh[32 + lane] = share[lane];
  __syncthreads();

  // ---- generation MLP: pre1 = W1 x + b1 ; h = relu ; u0 = W2 h + b2 ----
  float pre = b1[lane];
  for (int j = 0; j < 64; ++j) pre = fmaf(W1[lane * 64 + j], xsh[j], pre);
  presh[lane] = pre;
  float hv = fmaxf(pre, 0.f);
  hsh[lane] = hv;
  __syncthreads();
  float uv = b2[lane];
  for (int j = 0; j < 32; ++j) uv = fmaf(W2[lane * 32 + j], hsh[j], uv);
  ush[lane] = uv;
  __syncthreads();

  // ---- 5 local GD steps (E_s reused from LDS each step) ----
  for (int e = 0; e < 5; ++e) {
    wmma_z(Es, SPAD / 16, ush, zsh, lane);
    __syncthreads();
    for (int s = lane; s < NS; s += 32) gsh[s] = sigm(zsh[s]) - ysup[s];
    __syncthreads();
    float n2 = 0.f;
    for (int k = 0; k < 32; ++k) { float v = ush[k]; n2 = fmaf(v, v, n2); }
    const float nrm = sqrtf(n2);
    float g = 0.f;  // gu[lane] = sum_s g[s]*Es[s][lane]
    for (int s = 0; s < NS; ++s) g = fmaf(gsh[s], Es[s * PITCH + lane], g);
    float un = ush[lane];
    un = un - LOCAL_LR * (g + LAM_U * un / nrm);
    __syncthreads();
    ush[lane] = un;
    __syncthreads();
  }

  // ---- query pass ----
  wmma_z(Eq, QPAD / 16, ush, zsh, lane);
  __syncthreads();
  float lp = 0.f;
  for (int q = lane; q < NQ; q += 32) {
    float z = zsh[q];
    float y = yqry[q];
    float sp = fmaxf(z, 0.f) + log1pf(expf(-fabsf(z)));  // softplus, stable
    lp += sp - y * z;
    gsh[q] = sigm(z) - y;
  }
  red[lane] = lp;
  __syncthreads();
  float n2 = 0.f;
  for (int k = 0; k < 32; ++k) { float v = ush[k]; n2 = fmaf(v, v, n2); }
  const float nrm = sqrtf(n2);
  float tmpd = LAM_U * ush[lane] / nrm;
  for (int q = 0; q < NQ; ++q) tmpd = fmaf(gsh[q], Eq[q * PITCH + lane], tmpd);
  tmpsh[lane] = tmpd;
  tmp_ws[t * 32 + lane] = tmpd;
  __syncthreads();
  float dp = 0.f;  // dpre1 = (tmp @ W2) * (pre1 > 0)
  for (int j = 0; j < 32; ++j) dp = fmaf(tmpsh[j], W2[j * 32 + lane], dp);
  dp = (presh[lane] > 0.f) ? dp : 0.f;
  dpre1_ws[t * 32 + lane] = dp;
  h_ws[t * 32 + lane] = hsh[lane];
  x_ws[t * 64 + lane] = xsh[lane];
  x_ws[t * 64 + 32 + lane] = xsh[32 + lane];
  if (lane == 0) {
    float ls = 0.f;
    for (int k = 0; k < 32; ++k) ls += red[k];
    loss_ws[t] = ls + LAM_U * nrm;
  }
}

// Kernel 2: weight-grad GEMM reductions via WMMA.
// dW2[i][j] = sum_t tmp[t][i]*h[t][j]   (tiles 0..3,  ldb=32)
// dW1[i][j] = sum_t dpre1[t][i]*x[t][j] (tiles 4..11, ldb=64)
// One wave per (tile, K-slice of 512 tasks); partials to ws, combined in k3.
__global__ void __launch_bounds__(32)
k2_wgrad(const float* __restrict__ tmp_ws, const float* __restrict__ h_ws,
         const float* __restrict__ dpre1_ws, const float* __restrict__ x_ws,
         float* __restrict__ partials) {
  const int lane = threadIdx.x;
  const int tile = blockIdx.x % 12;
  const int ks = blockIdx.x / 12;  // 0..7
  const int l15 = lane & 15;
  const int koffx = (lane >= 16) ? 2 : 0;
  const int koffy = (lane >= 16) ? 3 : 1;

  const float* A; const float* B; int mi, nj, ldb;
  if (tile < 4) { A = tmp_ws;   B = h_ws; mi = tile >> 1;      nj = tile & 1;      ldb = 32; }
  else          { A = dpre1_ws; B = x_ws; mi = (tile - 4) >> 2; nj = (tile - 4) & 3; ldb = 64; }

  v8f c = {0.f, 0.f, 0.f, 0.f, 0.f, 0.f, 0.f, 0.f};
  const int t0base = ks * 512;
  for (int t0 = t0base; t0 < t0base + 512; t0 += 4) {
    v2f a, b;
    // A^T fragment: A[M=i][K=t] = src[t*32 + i]
    a.x = A[(t0 + koffx) * 32 + mi * 16 + l15];
    a.y = A[(t0 + koffy) * 32 + mi * 16 + l15];
    // B fragment: B[K=t][N=j] = src[t*ldb + j]
    b.x = B[(t0 + koffx) * ldb + nj * 16 + l15];
    b.y = B[(t0 + koffy) * ldb + nj * 16 + l15];
    c = __builtin_amdgcn_wmma_f32_16x16x4_f32(false, a, false, b, (short)0, c,
                                              false, false);
  }
  float* dst = partials + (tile * 8 + ks) * 256;  // 16x16 tile, M-major
  const int mbase = (lane >= 16) ? 8 : 0;
#pragma unroll
  for (int j = 0; j < 8; ++j) dst[(mbase + j) * 16 + l15] = c[j];
}

// Kernel 3: deterministic final reductions + output assembly.
// out layout: [0]=loss, [1..2048]=dW1, [2049..2080]=db1, [2081..3104]=dW2,
//             [3105..3136]=db2, [3137..3168]=dshare
__global__ void __launch_bounds__(256)
k3_finalize(const float* __restrict__ loss_ws, const float* __restrict__ tmp_ws,
            const float* __restrict__ dpre1_ws, const float* __restrict__ partials,
            const float* __restrict__ W1, float* __restrict__ out) {
  const int tid = threadIdx.x;
  __shared__ float sred[256];
  __shared__ float spart[8][32];
  __shared__ float sdb1[32];

  // loss_sum
  float lp = 0.f;
  for (int i = tid; i < NT; i += 256) lp += loss_ws[i];
  sred[tid] = lp;
  __syncthreads();
  for (int off = 128; off > 0; off >>= 1) {
    if (tid < off) sred[tid] += sred[tid + off];
    __syncthreads();
  }
  if (tid == 0) out[0] = sred[0];

  // db2 = sum_t tmp[t]
  {
    const int c = tid & 31, ch = tid >> 5;
    float a = 0.f;
    for (int t = ch * 512; t < ch * 512 + 512; ++t) a += tmp_ws[t * 32 + c];
    spart[ch][c] = a;
    __syncthreads();
    if (tid < 32) {
      float s = 0.f;
      for (int j = 0; j < 8; ++j) s += spart[j][tid];
      out[3105 + tid] = s;
    }
    __syncthreads();
  }
  // db1 = sum_t dpre1[t] (also kept for dshare)
  {
    const int c = tid & 31, ch = tid >> 5;
    float a = 0.f;
    for (int t = ch * 512; t < ch * 512 + 512; ++t) a += dpre1_ws[t * 32 + c];
    spart[ch][c] = a;
    __syncthreads();
    if (tid < 32) {
      float s = 0.f;
      for (int j = 0; j < 8; ++j) s += spart[j][tid];
      out[2049 + tid] = s;
      sdb1[tid] = s;
    }
    __syncthreads();
  }
  // dW2 [32x32] from tiles 0..3
  for (int e = tid; e < 1024; e += 256) {
    const int i = e >> 5, j = e & 31;
    const int tile = ((i >> 4) << 1) + (j >> 4);
    const int idx = ((i & 15) << 4) + (j & 15);
    float s = 0.f;
    for (int ksl = 0; ksl < 8; ++ksl) s += partials[(tile * 8 + ksl) * 256 + idx];
    out[2081 + e] = s;
  }
  // dW1 [32x64] from tiles 4..11
  for (int e = tid; e < 2048; e += 256) {
    const int i = e >> 6, j = e & 63;
    const int tile = 4 + ((i >> 4) << 2) + (j >> 4);
    const int idx = ((i & 15) << 4) + (j & 15);
    float s = 0.f;
    for (int ksl = 0; ksl < 8; ++ksl) s += partials[(tile * 8 + ksl) * 256 + idx];
    out[1 + e] = s;
  }
  // dshare = db1 @ W1[:, 32:]
  if (tid < 32) {
    float s = 0.f;
    for (int i = 0; i < 32; ++i) s = fmaf(sdb1[i], W1[i * 64 + 32 + tid], s);
    out[3137 + tid] = s;
  }
}

extern "C" void kernel_launch(void* const* d_in, const int* in_sizes, int n_in,
                              void* d_out, int out_size, void* d_ws, size_t ws_size,
                              hipStream_t stream) {
  const float* emb     = (const float*)d_in[0];
  const float* W1      = (const float*)d_in[1];
  const float* b1      = (const float*)d_in[2];
  const float* W2      = (const float*)d_in[3];
  const float* b2      = (const float*)d_in[4];
  const float* share   = (const float*)d_in[5];
  const float* supp_y  = (const float*)d_in[6];
  const float* query_y = (const float*)d_in[7];
  const int* supp_itms  = (const int*)d_in[8];
  const int* query_itms = (const int*)d_in[9];
  float* out = (float*)d_out;

  // Workspace carve-up (floats): ~2.74 MB total
  float* ws       = (float*)d_ws;
  float* h_ws     = ws;                   // NT*32
  float* x_ws     = h_ws + NT * 32;       // NT*64
  float* tmp_ws   = x_ws + NT * 64;       // NT*32
  float* dpre1_ws = tmp_ws + NT * 32;     // NT*32
  float* loss_ws  = dpre1_ws + NT * 32;   // NT
  float* partials = loss_ws + NT;         // 12*8*256

  k1_per_task<<<NT, 32, 0, stream>>>(emb, W1, b1, W2, b2, share, supp_y, query_y,
                                     supp_itms, query_itms, h_ws, x_ws, tmp_ws,
                                     dpre1_ws, loss_ws);
  k2_wgrad<<<96, 32, 0, stream>>>(tmp_ws, h_ws, dpre1_ws, x_ws, partials);
  k3_finalize<<<1, 256, 0, stream>>>(loss_ws, tmp_ws, dpre1_ws, partials, W1, out);
}